// MambaEncoder_61529701483007
// MI455X (gfx1250) — compile-verified
//
#include <hip/hip_runtime.h>
#include <cmath>

// Problem constants (from reference)
#define B_   16
#define L_   1024
#define DIN_ 64
#define DM_  256
#define NL_  4
#define DI_  512
#define DS_  16
#define DC_  4
#define DTR_ 16
#define M_   (B_ * L_)   // 16384 token rows

typedef __attribute__((ext_vector_type(2))) float v2f;
typedef __attribute__((ext_vector_type(8))) float v8f;
typedef __attribute__((ext_vector_type(4))) unsigned int v4u;
typedef __attribute__((ext_vector_type(8))) int v8i;
typedef __attribute__((ext_vector_type(4))) int v4i;

// ---------------------------------------------------------------------------
// TDM: issue a 2D tile load Global -> LDS (one call per wave; EXEC ignored).
// Descriptor per CDNA5 ISA §8.3-8.6. The tile itself is declared as the whole
// "tensor" (always in-bounds). LDS pad feature inserts 2 dwords every 32
// dwords -> LDS row stride = 34 floats (8B aligned, conflict-free ds_load_b64).
// ---------------------------------------------------------------------------
#define KC 32           // K-chunk (tile_dim0, contiguous floats)
#define LDSTR 34        // LDS row stride in floats (KC + 2 pad)

__device__ __forceinline__ void tdm_load_tile(unsigned lds_byte_off,
                                              const float* gsrc,
                                              int tile_rows, int row_stride) {
  const unsigned long long ga = (unsigned long long)(uintptr_t)gsrc;
  v4u g0;
  g0[0] = 1u;                                   // count=1 (valid), user mode
  g0[1] = lds_byte_off;                         // lds_addr
  g0[2] = (unsigned)(ga & 0xffffffffu);         // global_addr[31:0]
  g0[3] = (unsigned)((ga >> 32) & 0x1ffffffu)   // global_addr[56:32]
          | (2u << 30);                         // type = 2 ("image")
  const unsigned d0 = (unsigned)KC;             // tensor_dim0 = tile cols
  const unsigned d1 = (unsigned)tile_rows;      // tensor_dim1 = tile rows
  const unsigned long long s0 = (unsigned long long)row_stride;
  v8i g1;
  g1[0] = (int)((2u << 16)      // data_size = 2 -> 4 bytes
                | (1u << 20)    // pad_enable
                | (4u << 22)    // pad_interval = 4 -> every 32 dwords
                | (1u << 25));  // pad_amount = 1 -> 2 dwords
  g1[1] = (int)((d0 & 0xffffu) << 16);                         // td0[15:0]
  g1[2] = (int)(((d0 >> 16) & 0xffffu) | ((d1 & 0xffffu) << 16));
  g1[3] = (int)(((d1 >> 16) & 0xffffu) | ((unsigned)KC << 16)); // tile_dim0
  g1[4] = (int)((unsigned)tile_rows & 0xffffu);                 // tile_dim1
  g1[5] = (int)(unsigned)(s0 & 0xffffffffu);                    // d0_stride lo
  g1[6] = (int)(unsigned)((s0 >> 32) & 0xffffu);                // d0_stride hi
  g1[7] = 0;
  const v4i z4 = {};
#if __clang_major__ >= 23
  const v8i z8 = {};
  __builtin_amdgcn_tensor_load_to_lds(g0, g1, z4, z4, z8, 0);
#else
  __builtin_amdgcn_tensor_load_to_lds(g0, g1, z4, z4, 0);
#endif
}

// ---------------------------------------------------------------------------
// TDM + LDS + WMMA GEMM:  C[M,N] = A[M,K] @ W[N,K]^T + bias[N]
// Workgroup (8 waves) computes a 64x128 tile. Double-buffered LDS tiles:
// A 64xKC, W 128xKC, fetched by the Tensor Data Mover (TENSORcnt) so the
// next chunk's DMA overlaps this chunk's WMMAs. Wave w -> 16x64 sub-strip.
// Requires M%64==0, N%128==0, K%KC==0.
// ---------------------------------------------------------------------------
__global__ __launch_bounds__(256) void wmma_gemm_tdm_kernel(
    const float* __restrict__ A, int lda,
    const float* __restrict__ W, int ldw,
    const float* __restrict__ bias,
    float* __restrict__ C, int ldc, int K) {
  __shared__ float sA[2][64 * LDSTR];
  __shared__ float sW[2][128 * LDSTR];

  const int m0 = blockIdx.y * 64;
  const int n0 = blockIdx.x * 128;
  const int wid = threadIdx.x >> 5;
  const int lane = threadIdx.x & 31;
  const int hh = lane >> 4, r = lane & 15;
  const int wm = (wid & 3) << 4;   // wave M offset within block (0..48)
  const int wn = (wid >> 2) << 6;  // wave N offset within block (0 or 64)

  v8f acc[4] = {};

  if (threadIdx.x < 32) {  // wave 0 drives the TDM
    tdm_load_tile((unsigned)(uintptr_t)&sA[0][0], A + (size_t)m0 * lda, 64, lda);
    tdm_load_tile((unsigned)(uintptr_t)&sW[0][0], W + (size_t)n0 * ldw, 128, ldw);
  }

  const int nk = K / KC;
  for (int kc = 0; kc < nk; ++kc) {
    __builtin_amdgcn_s_wait_tensorcnt((short)0);  // chunk kc landed in LDS
    __syncthreads();
    if (kc + 1 < nk && threadIdx.x < 32) {        // prefetch chunk kc+1
      const int k1 = (kc + 1) * KC;
      tdm_load_tile((unsigned)(uintptr_t)&sA[(kc + 1) & 1][0],
                    A + (size_t)m0 * lda + k1, 64, lda);
      tdm_load_tile((unsigned)(uintptr_t)&sW[(kc + 1) & 1][0],
                    W + (size_t)n0 * ldw + k1, 128, ldw);
    }
    const float* la = sA[kc & 1];
    const float* lw = sW[kc & 1];
#pragma unroll
    for (int kk = 0; kk < KC; kk += 4) {
      v2f a;
      const int ao = (wm + r) * LDSTR + kk + 2 * hh;
      a[0] = la[ao];
      a[1] = la[ao + 1];
#pragma unroll
      for (int t = 0; t < 4; ++t) {
        const int bo = (wn + 16 * t + r) * LDSTR + kk + 2 * hh;
        v2f b;
        b[0] = lw[bo];
        b[1] = lw[bo + 1];
        acc[t] = __builtin_amdgcn_wmma_f32_16x16x4_f32(
            false, a, false, b, (short)0, acc[t], false, false);
      }
    }
    // next-iteration top barrier guarantees all waves finished reading this
    // buffer before the TDM refills it.
  }

#pragma unroll
  for (int t = 0; t < 4; ++t) {
    const int nc = n0 + wn + 16 * t + r;
    const float bv = bias ? bias[nc] : 0.0f;
#pragma unroll
    for (int v = 0; v < 8; ++v) {
      const int mr = m0 + wm + v + 8 * hh;
      C[(size_t)mr * ldc + nc] = acc[t][v] + bv;
    }
  }
}

// ---------------------------------------------------------------------------
// Register-direct WMMA GEMM (small shapes), software-pipelined fragments.
// ACT: 0 = none, 1 = softplus (dt projection).
// ---------------------------------------------------------------------------
__device__ __forceinline__ v2f ldfrag(const float* __restrict__ p, int ld,
                                      int row0, int k0, int r, int hh) {
  const float* q = p + (size_t)(row0 + r) * ld + (k0 + 2 * hh);
  v2f x;
  x[0] = q[0];
  x[1] = q[1];
  return x;
}

template <int TN, int ACT>
__global__ __launch_bounds__(256) void wmma_gemm_kernel(
    const float* __restrict__ A, int lda,
    const float* __restrict__ W, int ldw,
    const float* __restrict__ bias,
    float* __restrict__ C, int ldc,
    int Mtiles, int Nstrips, int K) {
  const int wid = blockIdx.x * (blockDim.x >> 5) + (threadIdx.x >> 5);
  if (wid >= Mtiles * Nstrips) return;  // wave-uniform exit (EXEC stays full)

  const int lane = threadIdx.x & 31;
  const int hh = lane >> 4, r = lane & 15;
  const int mt = wid / Nstrips, ns = wid % Nstrips;
  const int m0 = mt << 4, n0 = ns * (16 * TN);

  v8f acc[TN] = {};

  v2f a = ldfrag(A, lda, m0, 0, r, hh);
  v2f bb[TN];
#pragma unroll
  for (int t = 0; t < TN; ++t) bb[t] = ldfrag(W, ldw, n0 + 16 * t, 0, r, hh);

  for (int k0 = 0; k0 < K; k0 += 4) {
    v2f an = a;
    v2f bn[TN];
#pragma unroll
    for (int t = 0; t < TN; ++t) bn[t] = bb[t];
    const int k1 = k0 + 4;
    if (k1 < K) {  // issue next-step loads before this step's WMMAs
      an = ldfrag(A, lda, m0, k1, r, hh);
#pragma unroll
      for (int t = 0; t < TN; ++t) bn[t] = ldfrag(W, ldw, n0 + 16 * t, k1, r, hh);
    }
#pragma unroll
    for (int t = 0; t < TN; ++t)
      acc[t] = __builtin_amdgcn_wmma_f32_16x16x4_f32(
          false, a, false, bb[t], (short)0, acc[t], false, false);
    a = an;
#pragma unroll
    for (int t = 0; t < TN; ++t) bb[t] = bn[t];
  }

#pragma unroll
  for (int t = 0; t < TN; ++t) {
    const int nc = n0 + 16 * t + r;
    const float bv = bias ? bias[nc] : 0.0f;
#pragma unroll
    for (int v = 0; v < 8; ++v) {
      const int mr = m0 + v + 8 * hh;
      float val = acc[t][v] + bv;
      if (ACT == 1) val = (val > 20.0f) ? val : log1pf(__expf(val));
      C[(size_t)mr * ldc + nc] = val;
    }
  }
}

// ---------------------------------------------------------------------------
// Causal depthwise conv (DC=4) + bias + SiLU.
// ---------------------------------------------------------------------------
__global__ __launch_bounds__(256) void conv_silu_kernel(
    const float* __restrict__ xz, const float* __restrict__ cw,
    const float* __restrict__ cb, float* __restrict__ u) {
  const size_t idx = (size_t)blockIdx.x * blockDim.x + threadIdx.x;
  if (idx >= (size_t)M_ * DI_) return;
  const int d = (int)(idx % DI_);
  const size_t bl = idx / DI_;
  const int l = (int)(bl % L_);
  const size_t b = bl / L_;

  const float* col = xz + (b * L_) * (size_t)(2 * DI_) + d;
  float acc = cb[d];
#pragma unroll
  for (int j = 0; j < DC_; ++j) {
    const int ls = l - (DC_ - 1) + j;
    if (ls >= 0) acc += cw[d * DC_ + j] * col[(size_t)ls * (2 * DI_)];
  }
  u[idx] = acc / (1.0f + __expf(-acc));  // silu
}

// ---------------------------------------------------------------------------
// Selective scan: one thread per (b,d); 16-wide SSM state in registers.
// Fuses y = h.C + u*D and the silu(z) gate; writes y in place over u.
// ---------------------------------------------------------------------------
__global__ __launch_bounds__(256) void scan_kernel(
    const float* __restrict__ dt, const float* __restrict__ xdbl,
    const float* __restrict__ xz, const float* __restrict__ A_log,
    const float* __restrict__ Dp, float* uy) {
  const int d = blockIdx.x * blockDim.x + threadIdx.x;
  const int b = blockIdx.y;

  float Aa[DS_];
#pragma unroll
  for (int s = 0; s < DS_; ++s) Aa[s] = -__expf(A_log[(size_t)d * DS_ + s]);
  const float Dd = Dp[d];

  float hst[DS_];
#pragma unroll
  for (int s = 0; s < DS_; ++s) hst[s] = 0.0f;

  for (int t = 0; t < L_; ++t) {
    const size_t row = (size_t)b * L_ + t;
    const float dtv = dt[row * DI_ + d];
    const float uv  = uy[row * DI_ + d];
    const float zv  = xz[row * (size_t)(2 * DI_) + DI_ + d];
    const float* bc = xdbl + row * 48;

    float yv = 0.0f;
#pragma unroll
    for (int s = 0; s < DS_; ++s) {
      const float dA = __expf(dtv * Aa[s]);
      hst[s] = hst[s] * dA + (dtv * bc[DTR_ + s]) * uv;
      yv += hst[s] * bc[DTR_ + DS_ + s];
    }
    yv += uv * Dd;
    const float sz = zv / (1.0f + __expf(-zv));
    uy[row * DI_ + d] = yv * sz;

    if (t + 1 < L_) {  // global_prefetch_b8 for the next timestep
      __builtin_prefetch(dt + (row + 1) * DI_ + d, 0, 1);
      __builtin_prefetch(uy + (row + 1) * DI_ + d, 0, 1);
      __builtin_prefetch(xdbl + (row + 1) * 48, 0, 1);
    }
  }
}

// ---------------------------------------------------------------------------
// Residual add + LayerNorm over DM=256; one block per row; LDS tree reduce.
// ---------------------------------------------------------------------------
__global__ __launch_bounds__(256) void add_ln_kernel(
    const float* __restrict__ xin, const float* res,
    const float* __restrict__ g, const float* __restrict__ bb, float* dst) {
  __shared__ float red[DM_];
  const int row = blockIdx.x;
  const int c = threadIdx.x;
  const size_t idx = (size_t)row * DM_ + c;
  const float v = xin[idx] + res[idx];

  red[c] = v;
  __syncthreads();
  for (int off = DM_ / 2; off > 0; off >>= 1) {
    if (c < off) red[c] += red[c + off];
    __syncthreads();
  }
  const float mu = red[0] * (1.0f / DM_);
  __syncthreads();

  const float dv = v - mu;
  red[c] = dv * dv;
  __syncthreads();
  for (int off = DM_ / 2; off > 0; off >>= 1) {
    if (c < off) red[c] += red[c + off];
    __syncthreads();
  }
  const float var = red[0] * (1.0f / DM_);

  dst[idx] = dv * rsqrtf(var + 1e-5f) * g[c] + bb[c];
}

// ---------------------------------------------------------------------------
extern "C" void kernel_launch(void* const* d_in, const int* in_sizes, int n_in,
                              void* d_out, int out_size, void* d_ws,
                              size_t ws_size, hipStream_t stream) {
  (void)in_sizes; (void)n_in; (void)out_size; (void)ws_size;

  const float* x    = (const float*)d_in[0];
  const float* W_in = (const float*)d_in[1];
  const float* b_in = (const float*)d_in[2];
  const float* inw  = (const float*)d_in[3];
  const float* cw   = (const float*)d_in[4];
  const float* cb   = (const float*)d_in[5];
  const float* xpw  = (const float*)d_in[6];
  const float* dtw  = (const float*)d_in[7];
  const float* dtb  = (const float*)d_in[8];
  const float* Alog = (const float*)d_in[9];
  const float* Dp   = (const float*)d_in[10];
  const float* ow   = (const float*)d_in[11];
  const float* lng  = (const float*)d_in[12];
  const float* lnb  = (const float*)d_in[13];
  float* out = (float*)d_out;

  // Workspace layout (floats). tmp reuses xz (z consumed by scan first).
  float* ws   = (float*)d_ws;
  float* xz   = ws;                          // M * 1024
  float* u    = xz   + (size_t)M_ * 1024;    // M * 512  (y written in place)
  float* xdbl = u    + (size_t)M_ * 512;     // M * 48
  float* dt   = xdbl + (size_t)M_ * 48;      // M * 512
  float* h    = dt   + (size_t)M_ * 512;     // M * 256
  float* tmp  = xz;                          // M * 256 (reuse)

  const int Mt = M_ / 16;

  // h = x @ W_in^T + b_in          (N=256, K=64)  -- TDM path
  wmma_gemm_tdm_kernel<<<dim3(DM_ / 128, M_ / 64), 256, 0, stream>>>(
      x, DIN_, W_in, DIN_, b_in, h, DM_, DIN_);

  for (int i = 0; i < NL_; ++i) {
    const float* inw_i = inw + (size_t)i * (2 * DI_) * DM_;
    const float* cw_i  = cw  + (size_t)i * DI_ * DC_;
    const float* cb_i  = cb  + (size_t)i * DI_;
    const float* xpw_i = xpw + (size_t)i * (DTR_ + 2 * DS_) * DI_;
    const float* dtw_i = dtw + (size_t)i * DI_ * DTR_;
    const float* dtb_i = dtb + (size_t)i * DI_;
    const float* Al_i  = Alog + (size_t)i * DI_ * DS_;
    const float* Dp_i  = Dp  + (size_t)i * DI_;
    const float* ow_i  = ow  + (size_t)i * DM_ * DI_;
    const float* lng_i = lng + (size_t)i * DM_;
    const float* lnb_i = lnb + (size_t)i * DM_;

    // xz = h @ in_proj_w[i]^T      (N=1024, K=256) -- TDM path
    wmma_gemm_tdm_kernel<<<dim3((2 * DI_) / 128, M_ / 64), 256, 0, stream>>>(
        h, DM_, inw_i, DM_, nullptr, xz, 2 * DI_, DM_);

    // u = silu(causal_conv(xz[..., :DI]) + cb)
    conv_silu_kernel<<<dim3((unsigned)(((size_t)M_ * DI_) / 256)), 256, 0,
                       stream>>>(xz, cw_i, cb_i, u);

    // x_dbl = u @ x_proj_w[i]^T    (N=48, K=512) -- register path
    wmma_gemm_kernel<3, 0><<<dim3((Mt + 7) / 8), 256, 0, stream>>>(
        u, DI_, xpw_i, DI_, nullptr, xdbl, 48, Mt, 1, DI_);

    // dt = softplus(x_dbl[:, :16] @ dt_proj_w[i]^T + dt_proj_b[i]) (N=512,K=16)
    wmma_gemm_kernel<4, 1><<<dim3((Mt * (DI_ / 64) + 7) / 8), 256, 0, stream>>>(
        xdbl, 48, dtw_i, DTR_, dtb_i, dt, DI_, Mt, DI_ / 64, DTR_);

    // selective scan + gating, y written over u
    scan_kernel<<<dim3(DI_ / 256, B_), 256, 0, stream>>>(dt, xdbl, xz, Al_i,
                                                         Dp_i, u);

    // tmp = y @ out_w[i]^T         (N=256, K=512) -- TDM path
    wmma_gemm_tdm_kernel<<<dim3(DM_ / 128, M_ / 64), 256, 0, stream>>>(
        u, DI_, ow_i, DI_, nullptr, tmp, DM_, DI_);

    // h = LayerNorm(tmp + h) * g + b   (last layer -> d_out)
    float* dst = (i == NL_ - 1) ? out : h;
    add_ln_kernel<<<dim3(M_), 256, 0, stream>>>(tmp, h, lng_i, lnb_i, dst);
  }
}